// GammaCapsule_88708254531788
// MI455X (gfx1250) — compile-verified
//
#include <hip/hip_runtime.h>

typedef float v2f __attribute__((ext_vector_type(2)));
typedef float v8f __attribute__((ext_vector_type(8)));

#define B_ 64
#define O_ 32
#define I_ 2048
#define D_ 16
#define EPSF 1e-12f
// t_const = log(0.9*(O-1)) - log(1-0.9) = log(27.9) - log(0.1)
#define T_CONST 5.6312118f

// ---------------------------------------------------------------------------
// Per-wave micro-GEMM: for one (o,i), compute clipped u_hat for all 64 b.
// A = W[o,i,:,:]  (M=d, K=k, 16x16)      -> 4 K-chunks of 16x4
// B = u[:,i,:]^T  (K=k, N=b_local, 4x16) -> 4 b-tiles of 16
// D tile layout: lane L holds b = bt*16 + (L&15), d = 8*(L>>4) + v, v=0..7.
// Loads are batched up front so the 16 WMMAs issue without full loadcnt drains.
// ---------------------------------------------------------------------------
__device__ __forceinline__ void compute_uhat(const float* __restrict__ u,
                                             const float* __restrict__ W,
                                             const float* __restrict__ un,
                                             int o, int i, int lane,
                                             v8f acc[4])
{
    const int hl = lane >> 4;        // 0: K pair {0,1} / d 0..7 ; 1: K pair {2,3} / d 8..15
    const int lm = lane & 15;
    const float* Wp = W + (size_t)(o * I_ + i) * (D_ * D_) + lm * D_ + hl * 2;

    // ---- load phase: 4 A chunks + 16 B chunks, all issued before compute ----
    v2f a[4];
#pragma unroll
    for (int kc = 0; kc < 4; ++kc)
        a[kc] = *(const v2f*)(Wp + kc * 4);

    v2f bb[4][4];
#pragma unroll
    for (int bt = 0; bt < 4; ++bt) {
        const float* up = u + (size_t)((bt * 16 + lm) * I_ + i) * D_ + hl * 2;
#pragma unroll
        for (int kc = 0; kc < 4; ++kc)
            bb[bt][kc] = *(const v2f*)(up + kc * 4);
    }

    // ---- compute phase: 16 back-to-back WMMAs ----
    v8f z = {0.f, 0.f, 0.f, 0.f, 0.f, 0.f, 0.f, 0.f};
    acc[0] = z; acc[1] = z; acc[2] = z; acc[3] = z;
#pragma unroll
    for (int kc = 0; kc < 4; ++kc) {
#pragma unroll
        for (int bt = 0; bt < 4; ++bt) {
            acc[bt] = __builtin_amdgcn_wmma_f32_16x16x4_f32(
                false, a[kc], false, bb[bt][kc], (short)0, acc[bt], false, false);
        }
    }

    // ---- norm clip: u_hat *= min(|u_hat|, |u|) / (|u_hat| + EPS) ----
#pragma unroll
    for (int bt = 0; bt < 4; ++bt) {
        const int b = bt * 16 + lm;
        float ss = 0.f;
#pragma unroll
        for (int v = 0; v < 8; ++v) { float x = acc[bt][v]; ss += x * x; }
        ss += __shfl_xor(ss, 16, 32);             // combine d halves (lane pair L, L^16)
        float n = sqrtf(ss);
        float scale = fminf(n, un[b * I_ + i]) / (n + EPSF);
#pragma unroll
        for (int v = 0; v < 8; ++v) acc[bt][v] *= scale;
    }
}

// ---------------------------------------------------------------------------
// init: u_norm, zero d / s / scalars
// ---------------------------------------------------------------------------
__global__ void __launch_bounds__(256) init_kernel(const float* __restrict__ u,
                                                   float* un, float* dws,
                                                   float* s, float* scal)
{
    int idx = blockIdx.x * 256 + threadIdx.x;
    if (idx < B_ * O_ * I_) dws[idx] = 0.f;
    if (idx < B_ * I_) {
        const float* up = u + (size_t)idx * D_;
        float ss = 0.f;
#pragma unroll
        for (int k = 0; k < D_; ++k) ss += up[k] * up[k];
        un[idx] = sqrtf(ss);
    }
    if (idx < B_ * O_ * D_) s[idx] = 0.f;
    if (idx == 0) { scal[0] = 0.f; scal[1] = 0.f; }
}

// ---------------------------------------------------------------------------
// denom: per (b,i) softmax normalizers over O:  m = max(t*d), dn = sum exp(t*d-m)
// ---------------------------------------------------------------------------
__global__ void __launch_bounds__(256) denom_kernel(const float* __restrict__ dws,
                                                    const float* __restrict__ scal,
                                                    float* mws, float* dn)
{
    int idx = blockIdx.x * 256 + threadIdx.x;   // b*I + i
    if (idx >= B_ * I_) return;
    int b = idx / I_, i = idx % I_;
    float t = scal[0];
    float x[O_];
    float m = -1e30f;
#pragma unroll
    for (int o = 0; o < O_; ++o) {
        x[o] = t * dws[(size_t)(b * O_ + o) * I_ + i];
        m = fmaxf(m, x[o]);
    }
    float ssum = 0.f;
#pragma unroll
    for (int o = 0; o < O_; ++o) ssum += expf(x[o] - m);
    mws[idx] = m;
    dn[idx]  = ssum;
}

// ---------------------------------------------------------------------------
// s_kernel: recompute u_hat (WMMA), weight by c_ij, reduce over i into s[b,o,d]
// block = 8 waves = 8 consecutive i, one o; LDS tile [64x16] then global atomics
// ---------------------------------------------------------------------------
__global__ void __launch_bounds__(256) s_kernel(const float* __restrict__ u,
                                                const float* __restrict__ W,
                                                const float* __restrict__ un,
                                                const float* __restrict__ dws,
                                                const float* __restrict__ mws,
                                                const float* __restrict__ dn,
                                                const float* __restrict__ scal,
                                                float* s)
{
    __shared__ float tile[B_ * D_];   // 4 KB
    const int tid = threadIdx.x;
    for (int j = tid; j < B_ * D_; j += 256) tile[j] = 0.f;
    __syncthreads();

    const int wave = tid >> 5, lane = tid & 31;
    const int i = blockIdx.x * 8 + wave;
    const int o = blockIdx.y;
    const float t = scal[0];

    v8f acc[4];
    compute_uhat(u, W, un, o, i, lane, acc);

    const int hl = lane >> 4, lm = lane & 15, d0 = hl * 8;
#pragma unroll
    for (int bt = 0; bt < 4; ++bt) {
        const int b = bt * 16 + lm;
        float dv = dws[(size_t)(b * O_ + o) * I_ + i];
        float c = expf(t * dv - mws[b * I_ + i]) / dn[b * I_ + i];
#pragma unroll
        for (int v = 0; v < 8; ++v)
            atomicAdd(&tile[b * D_ + d0 + v], c * acc[bt][v]);
    }
    __syncthreads();

    for (int j = tid; j < B_ * D_; j += 256) {
        int b = j >> 4, d = j & 15;
        atomicAdd(&s[(b * O_ + o) * D_ + d], tile[j]);
    }
}

// ---------------------------------------------------------------------------
// squash: v = n2/(1+n2) * s/sqrt(n2+EPS); re-zero s for next iteration
// ---------------------------------------------------------------------------
__global__ void __launch_bounds__(256) squash_kernel(const float* __restrict__ bias,
                                                     float* s, float* vv,
                                                     float* out_v, int write_out)
{
    int idx = blockIdx.x * 256 + threadIdx.x;   // b*O + o
    if (idx >= B_ * O_) return;
    int o = idx % O_;
    float sv[D_]; float n2 = 0.f;
#pragma unroll
    for (int d = 0; d < D_; ++d) {
        float x = s[idx * D_ + d] + bias[o * D_ + d];
        sv[d] = x; n2 += x * x;
        s[idx * D_ + d] = 0.f;                  // ready for next s_kernel
    }
    float f = (n2 / (1.f + n2)) / sqrtf(n2 + EPSF);
#pragma unroll
    for (int d = 0; d < D_; ++d) {
        float val = f * sv[d];
        vv[idx * D_ + d] = val;
        if (write_out) out_v[idx * D_ + d] = val;
    }
}

// ---------------------------------------------------------------------------
// d_kernel: recompute u_hat (WMMA), d = ||v - u_hat||, store + global mean accum
// ---------------------------------------------------------------------------
__global__ void __launch_bounds__(256) d_kernel(const float* __restrict__ u,
                                                const float* __restrict__ W,
                                                const float* __restrict__ un,
                                                const float* __restrict__ vv,
                                                float* dws, float* scal)
{
    __shared__ float bsum;
    const int tid = threadIdx.x;
    if (tid == 0) bsum = 0.f;
    __syncthreads();

    const int wave = tid >> 5, lane = tid & 31;
    const int i = blockIdx.x * 8 + wave;
    const int o = blockIdx.y;

    v8f acc[4];
    compute_uhat(u, W, un, o, i, lane, acc);

    const int hl = lane >> 4, lm = lane & 15, d0 = hl * 8;
    float local = 0.f;
#pragma unroll
    for (int bt = 0; bt < 4; ++bt) {
        const int b = bt * 16 + lm;
        const float* vp = vv + (size_t)(b * O_ + o) * D_ + d0;
        float ss = 0.f;
#pragma unroll
        for (int v = 0; v < 8; ++v) { float x = vp[v] - acc[bt][v]; ss += x * x; }
        ss += __shfl_xor(ss, 16, 32);
        float dist = sqrtf(ss);
        if (hl == 0) {
            dws[(size_t)(b * O_ + o) * I_ + i] = dist;
            local += dist;
        }
    }
    atomicAdd(&bsum, local);
    __syncthreads();
    if (tid == 0) atomicAdd(&scal[1], bsum);
}

// ---------------------------------------------------------------------------
// t update: t = t_const / (d_p - d_o + 1e-12), d_p = 0.5*d_o
// ---------------------------------------------------------------------------
__global__ void t_kernel(float* scal)
{
    if (threadIdx.x == 0 && blockIdx.x == 0) {
        float d_o = scal[1] / (float)(B_ * O_ * I_);
        scal[0] = T_CONST / (-0.5f * d_o + 1e-12f);
        scal[1] = 0.f;
    }
}

// ---------------------------------------------------------------------------
// final c_ij output
// ---------------------------------------------------------------------------
__global__ void __launch_bounds__(256) cout_kernel(const float* __restrict__ dws,
                                                   const float* __restrict__ mws,
                                                   const float* __restrict__ dn,
                                                   const float* __restrict__ scal,
                                                   float* out_c)
{
    int idx = blockIdx.x * 256 + threadIdx.x;   // (b*O+o)*I + i
    if (idx >= B_ * O_ * I_) return;
    int i  = idx % I_;
    int b  = (idx / I_) / O_;
    float t = scal[0];
    out_c[idx] = expf(t * dws[idx] - mws[b * I_ + i]) / dn[b * I_ + i];
}

// ---------------------------------------------------------------------------
extern "C" void kernel_launch(void* const* d_in, const int* in_sizes, int n_in,
                              void* d_out, int out_size, void* d_ws, size_t ws_size,
                              hipStream_t stream) {
    (void)in_sizes; (void)n_in; (void)out_size; (void)ws_size;
    const float* u    = (const float*)d_in[0];   // [B,I,16]
    const float* W    = (const float*)d_in[1];   // [1,O,I,16,16]
    const float* bias = (const float*)d_in[2];   // [1,O,16]
    float* out = (float*)d_out;                  // v (B*O*16) ++ c (B*O*I)
    float* ws  = (float*)d_ws;

    float* un   = ws;                   // B*I
    float* dws  = un  + B_ * I_;        // B*O*I
    float* mws  = dws + B_ * O_ * I_;   // B*I
    float* dn   = mws + B_ * I_;        // B*I
    float* s    = dn  + B_ * I_;        // B*O*16
    float* vv   = s   + B_ * O_ * D_;   // B*O*16
    float* scal = vv  + B_ * O_ * D_;   // [t, dsum]

    const dim3 blk(256);
    const dim3 gemm_grid(I_ / 8, O_);

    init_kernel<<<(B_ * O_ * I_ + 255) / 256, blk, 0, stream>>>(u, un, dws, s, scal);

    for (int r = 0; r < 3; ++r) {
        denom_kernel<<<(B_ * I_ + 255) / 256, blk, 0, stream>>>(dws, scal, mws, dn);
        s_kernel<<<gemm_grid, blk, 0, stream>>>(u, W, un, dws, mws, dn, scal, s);
        squash_kernel<<<(B_ * O_ + 255) / 256, blk, 0, stream>>>(bias, s, vv, out,
                                                                 (r == 2) ? 1 : 0);
        if (r < 2) {
            d_kernel<<<gemm_grid, blk, 0, stream>>>(u, W, un, vv, dws, scal);
            t_kernel<<<1, 1, 0, stream>>>(scal);
        }
    }
    cout_kernel<<<(B_ * O_ * I_ + 255) / 256, blk, 0, stream>>>(dws, mws, dn, scal,
                                                                out + B_ * O_ * D_);
}